// GCNConv_1554778161247
// MI455X (gfx1250) — compile-verified
//
#include <hip/hip_runtime.h>
#include <hip/hip_bf16.h>

// ---------------------------------------------------------------------------
// GCNConv for MI455X (gfx1250, wave32).
//   h   = x @ W^T                  (fp32 WMMA 16x16x4, one 16x16 tile / wave)
//   out = b + sum_{e:(r->c)} dis[r]*dis[c]*h[r]  + dis[i]^2 * h[i]  (self loop)
// ---------------------------------------------------------------------------

#define N_NODES 10000
#define N_EDGES 640000
#define IN_C    128
#define OUT_C   128

typedef __attribute__((ext_vector_type(2))) float v2f;
typedef __attribute__((ext_vector_type(8))) float v8f;

// --- 1. deg[i] = 1.0 (self loop contribution) ------------------------------
__global__ void k_init_deg(float* __restrict__ deg) {
    int i = blockIdx.x * blockDim.x + threadIdx.x;
    if (i < N_NODES) deg[i] = 1.0f;
}

// --- 2. deg[col[e]] += 1 ----------------------------------------------------
__global__ void k_deg_accum(const int* __restrict__ col, float* __restrict__ deg) {
    int e = blockIdx.x * blockDim.x + threadIdx.x;
    if (e < N_EDGES) atomicAdd(&deg[col[e]], 1.0f);
}

// --- 3. dis[i] = rsqrt(deg[i]) ---------------------------------------------
__global__ void k_inv_sqrt(const float* __restrict__ deg, float* __restrict__ dis) {
    int i = blockIdx.x * blockDim.x + threadIdx.x;
    if (i < N_NODES) {
        float d = deg[i];
        dis[i] = (d > 0.0f) ? rsqrtf(d) : 0.0f;
    }
}

// --- 4. h = x @ W^T via V_WMMA_F32_16X16X4_F32 ------------------------------
// One wave per 16x16 output tile. Tiles: (10000/16)=625 x (128/16)=8 = 5000.
// Launch: 625 blocks x 256 threads (8 waves/block) -> exactly 5000 waves,
// so EXEC is all-1s for every wave (WMMA requirement).
__global__ void k_gemm_wmma(const float* __restrict__ x,
                            const float* __restrict__ W,
                            float* __restrict__ h) {
    const int wave   = (blockIdx.x * blockDim.x + threadIdx.x) >> 5;
    const int lane   = threadIdx.x & 31;
    const int tilesN = OUT_C / 16;           // 8
    const int tm     = wave / tilesN;        // 0..624
    const int tn     = wave % tilesN;        // 0..7

    const int mn    = lane & 15;             // row (A) / col (B) within tile
    const int khalf = (lane >> 4) << 1;      // 0 or 2: K sub-offset per lane half

    const float* __restrict__ xr = x + (tm * 16 + mn) * IN_C + khalf; // A: x[m, k..k+1]
    const float* __restrict__ wr = W + (tn * 16 + mn) * IN_C + khalf; // B: W[n, k..k+1] = B[k, n]

    v8f c = {};
    #pragma unroll 4
    for (int k = 0; k < IN_C; k += 4) {
        v2f a = *(const v2f*)(xr + k);
        v2f b = *(const v2f*)(wr + k);
        // 8 args: (neg_a, A, neg_b, B, c_mod, C, reuse_a, reuse_b)
        c = __builtin_amdgcn_wmma_f32_16x16x4_f32(false, a, false, b,
                                                  (short)0, c, false, false);
    }

    // D layout: VGPR v holds D[M = v + 8*(lane>>4), N = lane&15]
    const int mbase = tm * 16 + ((lane >> 4) << 3);
    const int ncol  = tn * 16 + mn;
    #pragma unroll
    for (int v = 0; v < 8; ++v) {
        h[(mbase + v) * OUT_C + ncol] = c[v];
    }
}

// --- 5. out = h * dis^2 + b  (self-loop term; also initializes d_out) ------
__global__ void k_out_init(const float* __restrict__ h,
                           const float* __restrict__ dis,
                           const float* __restrict__ bias,
                           float* __restrict__ out) {
    int i = blockIdx.x * blockDim.x + threadIdx.x;
    if (i < N_NODES * OUT_C) {
        int node = i >> 7;          // / 128
        int ch   = i & 127;
        float d  = dis[node];
        out[i] = h[i] * d * d + bias[ch];
    }
}

// --- 6. per-edge scatter: out[col] += h[row] * dis[row]*dis[col] ------------
// One wave per edge (grid-stride); lane owns 4 channels -> float4 gather +
// 4x global_atomic_add_f32. 32 lanes * 4 = 128 channels, fully coalesced.
__global__ void k_edge_scatter(const int* __restrict__ row,
                               const int* __restrict__ col,
                               const float* __restrict__ h,
                               const float* __restrict__ dis,
                               float* __restrict__ out) {
    const int wid    = (blockIdx.x * blockDim.x + threadIdx.x) >> 5;
    const int lane   = threadIdx.x & 31;
    const int nwaves = (gridDim.x * blockDim.x) >> 5;

    for (int e = wid; e < N_EDGES; e += nwaves) {
        const int r = row[e];
        const int c = col[e];
        const float norm = dis[r] * dis[c];

        float4 v = *(const float4*)(h + r * OUT_C + lane * 4);
        float* o = out + c * OUT_C + lane * 4;
        atomicAdd(o + 0, v.x * norm);
        atomicAdd(o + 1, v.y * norm);
        atomicAdd(o + 2, v.z * norm);
        atomicAdd(o + 3, v.w * norm);
    }
}

extern "C" void kernel_launch(void* const* d_in, const int* in_sizes, int n_in,
                              void* d_out, int out_size, void* d_ws, size_t ws_size,
                              hipStream_t stream) {
    const float* x  = (const float*)d_in[0];                 // [N_NODES, IN_C]
    const int*   ei = (const int*)  d_in[1];                 // [2, N_EDGES] flat
    const float* W  = (const float*)d_in[2];                 // [OUT_C, IN_C]
    const float* b  = (const float*)d_in[3];                 // [OUT_C]
    float*       out = (float*)d_out;                        // [N_NODES, OUT_C]

    const int* row = ei;                                     // edge_index[0]
    const int* col = ei + N_EDGES;                           // edge_index[1]

    // Workspace layout (all 16B aligned): deg | dis | h
    float* deg = (float*)d_ws;                               // N_NODES
    float* dis = deg + N_NODES;                              // N_NODES
    float* h   = dis + N_NODES;                              // N_NODES * OUT_C

    // 1. self-loop degree
    k_init_deg<<<(N_NODES + 255) / 256, 256, 0, stream>>>(deg);
    // 2. accumulate destination degrees
    k_deg_accum<<<(N_EDGES + 255) / 256, 256, 0, stream>>>(col, deg);
    // 3. deg^{-1/2}
    k_inv_sqrt<<<(N_NODES + 255) / 256, 256, 0, stream>>>(deg, dis);
    // 4. h = x @ W^T : 5000 tiles, 8 waves/block -> 625 blocks exactly
    k_gemm_wmma<<<625, 256, 0, stream>>>(x, W, h);
    // 5. out = h*dis^2 + b
    k_out_init<<<(N_NODES * OUT_C + 255) / 256, 256, 0, stream>>>(h, dis, b, out);
    // 6. edge scatter with f32 atomics (everything L2-resident)
    k_edge_scatter<<<4096, 256, 0, stream>>>(row, col, h, dis, out);
}